// FaceInpaintingHead_88622355185792
// MI455X (gfx1250) — compile-verified
//
#include <hip/hip_runtime.h>
#include <hip/hip_bf16.h>
#include <math.h>

#define B_        32
#define Hh        96
#define Ww_       96
#define DIMC      64
#define EXPC      256
#define PIX_PER_B (Hh*Ww_)        // 9216
#define NPIX      (B_*PIX_PER_B)  // 294912
#define KMASK     1152
#define LATENTD   1024

typedef float v2f __attribute__((ext_vector_type(2)));
typedef float v8f __attribute__((ext_vector_type(8)));

// ---- CDNA5 async global->LDS path (guarded; falls back to sync copies) ----
#if defined(__has_builtin)
#if __has_builtin(__builtin_amdgcn_global_load_async_to_lds_b128) && \
    __has_builtin(__builtin_amdgcn_s_wait_asynccnt)
#define HAS_ASYNC_LDS 1
#endif
#endif

#ifdef HAS_ASYNC_LDS
typedef int v4i __attribute__((vector_size(4 * sizeof(int))));
typedef __attribute__((address_space(1))) v4i* gv4i_p;   // global src
typedef __attribute__((address_space(3))) v4i* lv4i_p;   // LDS dst
__device__ __forceinline__ void async_b128(const void* g, void* l) {
  __builtin_amdgcn_global_load_async_to_lds_b128((gv4i_p)g, (lv4i_p)l, 0, 0);
}
#endif

__device__ __forceinline__ v8f wmma4(v2f a, v2f b, v8f c) {
  // D = A(16x4,f32) * B(4x16,f32) + C(16x16,f32) ; full fp32 matrix path on CDNA5
  return __builtin_amdgcn_wmma_f32_16x16x4_f32(false, a, false, b, (short)0, c, false, false);
}

__device__ __forceinline__ float gelu(float x) {
  return 0.5f * x * (1.0f + erff(x * 0.7071067811865476f));
}

// ---------------- latent projection: (32,1024)@(1024,64)+b, gelu -> lc ----------------
__global__ void __launch_bounds__(256) k_latent(const float* __restrict__ lat,
                                                const float* __restrict__ Wp,
                                                const float* __restrict__ bp,
                                                float* __restrict__ lc) {
  int tid = threadIdx.x, wave = tid >> 5, lane = tid & 31, lid = lane & 15, half = lane >> 4;
  int m0 = (wave & 1) * 16, n0 = (wave >> 1) * 16;   // 2 x 4 tiles, one per wave
  v8f c = {};
  const float* arow = lat + (size_t)(m0 + lid) * LATENTD + 2 * half;
  for (int k = 0; k < LATENTD; k += 4) {
    v2f a = *(const v2f*)(arow + k);
    v2f b;
    b.x = Wp[(size_t)(k + 2 * half) * DIMC + n0 + lid];
    b.y = Wp[(size_t)(k + 2 * half + 1) * DIMC + n0 + lid];
    c = wmma4(a, b, c);
  }
  float bias = bp[n0 + lid];
  for (int r = 0; r < 8; r++)
    lc[(m0 + r + 8 * half) * DIMC + n0 + lid] = gelu(c[r] + bias);
}

// ---------------- pre-contract latent channels of c1 into per-tap constants ----------------
__global__ void k_lc_taps(const float* __restrict__ lc, const float* __restrict__ c1w,
                          float* __restrict__ lct) {
  int b = blockIdx.x, t = threadIdx.x;          // t < 576 : o = t/9, tap = t%9
  int o = t / 9, tap = t % 9;
  float acc = 0.f;
  for (int i = 0; i < DIMC; i++) acc += lc[b * DIMC + i] * c1w[(o * 66 + 2 + i) * 9 + tap];
  lct[b * 576 + t] = acc;
}

// ---------------- c1: 3x3 conv (2 face ch + broadcast latent) -> NHWC x0, + GN stats ----------------
__global__ void __launch_bounds__(256) k_conv1(const float* __restrict__ face,
                                               const float* __restrict__ lct,
                                               const float* __restrict__ c1w,
                                               const float* __restrict__ c1b,
                                               float* __restrict__ x0,
                                               float* __restrict__ gst) {
  __shared__ float ssum[8], ssq[8];
  int tid = threadIdx.x;
  int gid = blockIdx.x * 256 + tid;
  int o = gid & 63; int p = gid >> 6;
  int w = p % Ww_; int h = (p / Ww_) % Hh; int b = p / PIX_PER_B;
  float acc = c1b[o];
  const float* f0 = face + (size_t)(b * 2 + 0) * PIX_PER_B;
  const float* f1 = face + (size_t)(b * 2 + 1) * PIX_PER_B;
  const float* lt = lct + b * 576 + o * 9;
  const float* wo = c1w + o * 66 * 9;
  for (int dy = 0; dy < 3; dy++) {
    int ih = h + dy - 1; if ((unsigned)ih >= (unsigned)Hh) continue;
    for (int dx = 0; dx < 3; dx++) {
      int iw = w + dx - 1; if ((unsigned)iw >= (unsigned)Ww_) continue;
      int tap = dy * 3 + dx; int off = ih * Ww_ + iw;
      acc += lt[tap] + f0[off] * wo[tap] + f1[off] * wo[9 + tap];
    }
  }
  x0[gid] = acc;
  if (tid < 8) { ssum[tid] = 0.f; ssq[tid] = 0.f; }
  __syncthreads();
  atomicAdd(&ssum[o >> 3], acc);
  atomicAdd(&ssq[o >> 3], acc * acc);
  __syncthreads();
  if (tid < 8) {
    atomicAdd(&gst[(b * 8 + tid) * 2 + 0], ssum[tid]);
    atomicAdd(&gst[(b * 8 + tid) * 2 + 1], ssq[tid]);
  }
}

__global__ void k_gn_fin(float* gst) {
  int t = threadIdx.x;                           // 256 = 32 b * 8 groups
  float n = 8.f * (float)PIX_PER_B;
  float mean = gst[t * 2] / n;
  float var  = gst[t * 2 + 1] / n - mean * mean;
  gst[t * 2]     = mean;
  gst[t * 2 + 1] = rsqrtf(var + 1e-5f);
}

__global__ void __launch_bounds__(256) k_gn_apply(float* __restrict__ x0,
                                                  const float* __restrict__ gst,
                                                  const float* __restrict__ g,
                                                  const float* __restrict__ bb) {
  int gid = blockIdx.x * 256 + threadIdx.x;
  int o = gid & 63; int b = gid / (PIX_PER_B * DIMC);
  int gi = (b * 8 + (o >> 3)) * 2;
  float v = (x0[gid] - gst[gi]) * gst[gi + 1] * g[o] + bb[o];
  x0[gid] = gelu(v);
}

// ---------------- depthwise 7x7 (NHWC, float4 over channels) ----------------
__global__ void k_prep_dww(const float* __restrict__ w, float* __restrict__ wT) {
  int idx = blockIdx.x * 256 + threadIdx.x;
  if (idx >= DIMC * 49) return;
  int c = idx / 49, tap = idx % 49;
  wT[tap * DIMC + c] = w[idx];
}

__global__ void __launch_bounds__(256) k_dwconv7(const float* __restrict__ in,
                                                 const float* __restrict__ wT,
                                                 const float* __restrict__ bias,
                                                 float* __restrict__ out) {
  int gid = blockIdx.x * 256 + threadIdx.x;
  int c4 = (gid & 15) * 4; int p = gid >> 4;
  int w = p % Ww_; int h = (p / Ww_) % Hh; int b = p / PIX_PER_B;
  float4 acc = *(const float4*)(bias + c4);
  for (int dy = 0; dy < 7; dy++) {
    int ih = h + dy - 3; if ((unsigned)ih >= (unsigned)Hh) continue;
    for (int dx = 0; dx < 7; dx++) {
      int iw = w + dx - 3; if ((unsigned)iw >= (unsigned)Ww_) continue;
      float4 xv = *(const float4*)(in + (size_t)((b * Hh + ih) * Ww_ + iw) * DIMC + c4);
      float4 wv = *(const float4*)(wT + (dy * 7 + dx) * DIMC + c4);
      acc.x += xv.x * wv.x; acc.y += xv.y * wv.y;
      acc.z += xv.z * wv.z; acc.w += xv.w * wv.w;
    }
  }
  *(float4*)(out + (size_t)p * DIMC + c4) = acc;
}

// ---------------- per-pixel LayerNorm over 64 channels (in place) ----------------
__global__ void __launch_bounds__(256) k_layernorm(float* __restrict__ io,
                                                   const float* __restrict__ g,
                                                   const float* __restrict__ bb) {
  int p = blockIdx.x * 256 + threadIdx.x;
  float v[64];
  float4* v4 = (float4*)v;
  const float4* src = (const float4*)(io + (size_t)p * DIMC);
  float sum = 0.f;
  for (int j = 0; j < 16; j++) {
    v4[j] = src[j];
    sum += v[4*j] + v[4*j+1] + v[4*j+2] + v[4*j+3];
  }
  float mean = sum * (1.f / 64.f);
  float var = 0.f;
  for (int i = 0; i < 64; i++) { float d = v[i] - mean; var += d * d; }
  float rstd = rsqrtf(var * (1.f / 64.f) + 1e-6f);
  for (int i = 0; i < 64; i++) v[i] = (v[i] - mean) * rstd * g[i] + bb[i];
  float4* dst = (float4*)(io + (size_t)p * DIMC);
  for (int j = 0; j < 16; j++) dst[j] = v4[j];
}

// ---------------- pw1: (NPIX,64)@(64,256)+b, gelu -> Y ; fused GRN sum(y^2) ----------------
__global__ void __launch_bounds__(256) k_gemm_pw1(const float* __restrict__ A,
                                                  const float* __restrict__ Bw,
                                                  const float* __restrict__ bias,
                                                  float* __restrict__ Y,
                                                  float* __restrict__ gx2) {
  __shared__ float As[64 * 68];
  __shared__ float Bs[64 * 68];
  const int m0 = blockIdx.x * 64;
  const int n0 = blockIdx.y * 64;
  const int tid = threadIdx.x;
  {
    int r = tid >> 2, c0 = (tid & 3) * 16;
    const float* ga = A + (size_t)(m0 + r) * DIMC + c0;
    const float* gw = Bw + (size_t)r * EXPC + n0 + c0;
    float* la = As + r * 68 + c0;
    float* lb = Bs + r * 68 + c0;
#ifdef HAS_ASYNC_LDS
    for (int j = 0; j < 4; j++) {
      async_b128(ga + 4 * j, la + 4 * j);
      async_b128(gw + 4 * j, lb + 4 * j);
    }
    __builtin_amdgcn_s_wait_asynccnt(0);
#else
    for (int j = 0; j < 4; j++) {
      ((float4*)la)[j] = ((const float4*)ga)[j];
      ((float4*)lb)[j] = ((const float4*)gw)[j];
    }
#endif
  }
  __syncthreads();
  int wave = tid >> 5, lane = tid & 31, lid = lane & 15, half = lane >> 4;
  int ms = (wave & 3) * 16, ns = (wave >> 2) * 32;
  v8f c0v = {}, c1v = {};
  const float* ap = As + (ms + lid) * 68 + 2 * half;
  for (int k = 0; k < 64; k += 4) {
    v2f a = *(const v2f*)(ap + k);
    const float* b0p = Bs + (k + 2 * half) * 68 + ns + lid;
    v2f b0, b1;
    b0.x = b0p[0];  b0.y = b0p[68];
    b1.x = b0p[16]; b1.y = b0p[84];
    c0v = wmma4(a, b0, c0v);
    c1v = wmma4(a, b1, c1v);
  }
  int b = m0 / PIX_PER_B;                   // tile never crosses batch (9216 % 64 == 0)
  int nA = n0 + ns + lid, nB = nA + 16;
  float bA = bias[nA], bB = bias[nB];
  float s0 = 0.f, s1 = 0.f;
  for (int r = 0; r < 8; r++) {
    size_t m = m0 + ms + r + 8 * half;
    float v0 = gelu(c0v[r] + bA);
    float v1 = gelu(c1v[r] + bB);
    Y[m * EXPC + nA] = v0;
    Y[m * EXPC + nB] = v1;
    s0 += v0 * v0; s1 += v1 * v1;
  }
  atomicAdd(&gx2[b * EXPC + nA], s0);
  atomicAdd(&gx2[b * EXPC + nB], s1);
}

// ---------------- GRN scale: s = 1 + g * gx / (mean_ch(gx)+1e-6) ----------------
__global__ void k_grn_scale(const float* __restrict__ gx2, const float* __restrict__ gg,
                            float* __restrict__ s) {
  __shared__ float red[256];
  int b = blockIdx.x, c = threadIdx.x;
  float gx = sqrtf(gx2[b * EXPC + c]);
  red[c] = gx; __syncthreads();
  for (int st = 128; st > 0; st >>= 1) { if (c < st) red[c] += red[c + st]; __syncthreads(); }
  float mean = red[0] * (1.f / (float)EXPC);
  s[b * EXPC + c] = 1.f + gg[c] * gx / (mean + 1e-6f);
}

// ---------------- pw2: (NPIX,256)@(256,64)+b + residual ----------------
// GRN fold (a = y*s + grn_b) applied at A-fragment load; K chunks double-buffered
// through LDS with CDNA5 async global->LDS DMA when available.
__global__ void __launch_bounds__(256) k_gemm_pw2(const float* __restrict__ Y,
                                                  const float* __restrict__ s,
                                                  const float* __restrict__ gb,
                                                  const float* __restrict__ Bw,
                                                  const float* __restrict__ bias,
                                                  const float* __restrict__ resid,
                                                  float* __restrict__ out) {
  __shared__ float As[2][64 * 68];
  __shared__ float Bs[2][64 * 68];
  const int m0 = blockIdx.x * 64;
  const int b = m0 / PIX_PER_B;
  const int tid = threadIdx.x;
  int wave = tid >> 5, lane = tid & 31, lid = lane & 15, half = lane >> 4;
  int ms = (wave & 3) * 16, ns = (wave >> 2) * 32;
  int r = tid >> 2, c0 = (tid & 3) * 16;
  v8f c0v = {}, c1v = {};

#ifdef HAS_ASYNC_LDS
  auto issue = [&](int kc, int buf) {
    const float* gy = Y + (size_t)(m0 + r) * EXPC + kc * 64 + c0;
    const float* gw = Bw + (size_t)(kc * 64 + r) * DIMC + c0;
    float* la = &As[buf][r * 68 + c0];
    float* lb = &Bs[buf][r * 68 + c0];
    for (int j = 0; j < 4; j++) {
      async_b128(gy + 4 * j, la + 4 * j);
      async_b128(gw + 4 * j, lb + 4 * j);
    }
  };
  issue(0, 0);
#endif

  for (int kc = 0; kc < 4; kc++) {
    const int buf = kc & 1;
#ifdef HAS_ASYNC_LDS
    if (kc < 3) { issue(kc + 1, buf ^ 1); __builtin_amdgcn_s_wait_asynccnt(8); }
    else        { __builtin_amdgcn_s_wait_asynccnt(0); }
#else
    {
      const float4* gy = (const float4*)(Y + (size_t)(m0 + r) * EXPC + kc * 64 + c0);
      const float4* gw = (const float4*)(Bw + (size_t)(kc * 64 + r) * DIMC + c0);
      float4* la = (float4*)&As[buf][r * 68 + c0];
      float4* lb = (float4*)&Bs[buf][r * 68 + c0];
      for (int j = 0; j < 4; j++) { la[j] = gy[j]; lb[j] = gw[j]; }
    }
#endif
    __syncthreads();
    const float* ap = &As[buf][(ms + lid) * 68 + 2 * half];
    const float* sp = s + b * EXPC + kc * 64 + 2 * half;
    const float* gp = gb + kc * 64 + 2 * half;
    for (int k = 0; k < 64; k += 4) {
      v2f a  = *(const v2f*)(ap + k);
      v2f sv = *(const v2f*)(sp + k);
      v2f gv = *(const v2f*)(gp + k);
      a = a * sv + gv;                       // GRN fold
      const float* b0p = &Bs[buf][(k + 2 * half) * 68 + ns + lid];
      v2f b0, b1;
      b0.x = b0p[0];  b0.y = b0p[68];
      b1.x = b0p[16]; b1.y = b0p[84];
      c0v = wmma4(a, b0, c0v);
      c1v = wmma4(a, b1, c1v);
    }
    __syncthreads();
  }
  int nA = ns + lid, nB = nA + 16;
  float bA = bias[nA], bB = bias[nB];
  for (int rr = 0; rr < 8; rr++) {
    size_t m = m0 + ms + rr + 8 * half;
    out[m * DIMC + nA] = resid[m * DIMC + nA] + c0v[rr] + bA;
    out[m * DIMC + nB] = resid[m * DIMC + nB] + c1v[rr] + bB;
  }
}

// ---------------- ph1: 3x3 conv 64->32 + gelu ----------------
__global__ void k_prep_ph1w(const float* __restrict__ w, float* __restrict__ wT) {
  int idx = blockIdx.x * 256 + threadIdx.x;   // < 32*64*9
  int o = idx / (64 * 9); int rem = idx % (64 * 9); int i = rem / 9; int tap = rem % 9;
  wT[(o * 9 + tap) * 64 + i] = w[idx];
}

__global__ void __launch_bounds__(256) k_conv_ph1(const float* __restrict__ in,
                                                  const float* __restrict__ wT,
                                                  const float* __restrict__ bias,
                                                  float* __restrict__ out) {
  int gid = blockIdx.x * 256 + threadIdx.x;
  int o = gid & 31; int p = gid >> 5;
  int w = p % Ww_; int h = (p / Ww_) % Hh; int b = p / PIX_PER_B;
  float acc = bias[o];
  for (int dy = 0; dy < 3; dy++) {
    int ih = h + dy - 1; if ((unsigned)ih >= (unsigned)Hh) continue;
    for (int dx = 0; dx < 3; dx++) {
      int iw = w + dx - 1; if ((unsigned)iw >= (unsigned)Ww_) continue;
      const float4* xp = (const float4*)(in + (size_t)((b * Hh + ih) * Ww_ + iw) * DIMC);
      const float4* wp = (const float4*)(wT + (o * 9 + dy * 3 + dx) * DIMC);
      for (int j = 0; j < 16; j++) {
        float4 x4 = xp[j], w4 = wp[j];
        acc += x4.x * w4.x + x4.y * w4.y + x4.z * w4.z + x4.w * w4.w;
      }
    }
  }
  out[(size_t)p * 32 + o] = gelu(acc);
}

// ---------------- ph2 (1x1, 32->2) computed ONLY at masked pixels + gather outputs ----------------
__global__ void k_ph2_gather(const float* __restrict__ ph1o, const int* __restrict__ midx,
                             const float* __restrict__ w, const float* __restrict__ bias,
                             float* __restrict__ out) {
  int gid = blockIdx.x * 256 + threadIdx.x;    // < B*K
  int k = gid % KMASK; int b = gid / KMASK;
  int pos = midx[b * KMASK + k];
  const float* xp = ph1o + ((size_t)b * PIX_PER_B + pos) * 32;
  float a0 = bias[0], a1 = bias[1];
  for (int i = 0; i < 32; i++) { float v = xp[i]; a0 += v * w[i]; a1 += v * w[32 + i]; }
  size_t o0 = (size_t)gid * 2;
  out[o0] = a0; out[o0 + 1] = a1;
  float* mi = out + (size_t)B_ * KMASK * 2;
  mi[o0]     = (float)(pos / Ww_);
  mi[o0 + 1] = (float)(pos % Ww_);
  float* vm = out + (size_t)B_ * KMASK * 4;
  vm[gid] = 1.0f;
}

extern "C" void kernel_launch(void* const* d_in, const int* in_sizes, int n_in,
                              void* d_out, int out_size, void* d_ws, size_t ws_size,
                              hipStream_t stream) {
  (void)in_sizes; (void)n_in; (void)out_size; (void)ws_size;
  const float* face   = (const float*)d_in[0];
  const float* latent = (const float*)d_in[1];
  const int*   midx   = (const int*)d_in[2];
  const float* lp_w = (const float*)d_in[3];
  const float* lp_b = (const float*)d_in[4];
  const float* c1_w = (const float*)d_in[5];
  const float* c1_b = (const float*)d_in[6];
  const float* gn_g = (const float*)d_in[7];
  const float* gn_b = (const float*)d_in[8];
  const float* blk[2][10];
  for (int bi = 0; bi < 2; bi++)
    for (int j = 0; j < 10; j++) blk[bi][j] = (const float*)d_in[9 + bi * 10 + j];
  const float* ph1_w = (const float*)d_in[29];
  const float* ph1_b = (const float*)d_in[30];
  const float* ph2_w = (const float*)d_in[31];
  const float* ph2_b = (const float*)d_in[32];

  float* ws = (float*)d_ws;
  float* lc   = ws;             // 2048
  float* lct  = ws + 2048;      // 18432
  float* gst  = ws + 20480;     // 512
  float* gx2a = ws + 20992;     // 8192
  float* gx2b = ws + 29184;     // 8192
  float* sbuf = ws + 37376;     // 8192
  float* dwT  = ws + 45568;     // 3136
  float* p1wT = ws + 48704;     // 18432
  float* x0   = ws + 67136;                          // NPIX*64
  float* xa   = x0 + (size_t)NPIX * DIMC;            // NPIX*64
  float* x1   = xa + (size_t)NPIX * DIMC;            // NPIX*64
  float* ybuf = x1 + (size_t)NPIX * DIMC;            // NPIX*256
  float* p1o  = ybuf + (size_t)NPIX * EXPC;          // NPIX*32
  float* out  = (float*)d_out;

  (void)hipMemsetAsync(gst, 0, 512 * sizeof(float), stream);
  (void)hipMemsetAsync(gx2a, 0, 8192 * sizeof(float), stream);
  (void)hipMemsetAsync(gx2b, 0, 8192 * sizeof(float), stream);

  k_latent<<<1, 256, 0, stream>>>(latent, lp_w, lp_b, lc);
  k_lc_taps<<<B_, 576, 0, stream>>>(lc, c1_w, lct);
  k_conv1<<<NPIX * DIMC / 256, 256, 0, stream>>>(face, lct, c1_w, c1_b, x0, gst);
  k_gn_fin<<<1, 256, 0, stream>>>(gst);
  k_gn_apply<<<NPIX * DIMC / 256, 256, 0, stream>>>(x0, gst, gn_g, gn_b);

  float* xin = x0; float* xout = x1;
  float* gx2 = gx2a;
  for (int bi = 0; bi < 2; bi++) {
    const float* const* P = blk[bi];
    k_prep_dww<<<(DIMC * 49 + 255) / 256, 256, 0, stream>>>(P[0], dwT);
    k_dwconv7<<<NPIX * 16 / 256, 256, 0, stream>>>(xin, dwT, P[1], xa);
    k_layernorm<<<NPIX / 256, 256, 0, stream>>>(xa, P[2], P[3]);
    k_gemm_pw1<<<dim3(NPIX / 64, 4), 256, 0, stream>>>(xa, P[4], P[5], ybuf, gx2);
    k_grn_scale<<<B_, 256, 0, stream>>>(gx2, P[6], sbuf);
    k_gemm_pw2<<<NPIX / 64, 256, 0, stream>>>(ybuf, sbuf, P[7], P[8], P[9], xin, xout);
    float* t = xin; xin = xout; xout = t;
    gx2 = gx2b;
  }
  // after two blocks xin holds the final feature map
  k_prep_ph1w<<<(32 * 64 * 9) / 256, 256, 0, stream>>>(ph1_w, p1wT);
  k_conv_ph1<<<NPIX * 32 / 256, 256, 0, stream>>>(xin, p1wT, ph1_b, p1o);
  k_ph2_gather<<<B_ * KMASK / 256, 256, 0, stream>>>(p1o, midx, ph2_w, ph2_b, out);
}